// SpreadEdgePool_11347303596515
// MI455X (gfx1250) — compile-verified
//
#include <hip/hip_runtime.h>
#include <math.h>

typedef float v2f __attribute__((ext_vector_type(2)));
typedef float v8f __attribute__((ext_vector_type(8)));

#define B_  4
#define N_  20000
#define C_  64
#define E_  320000
#define NK_ 10000

// ---------------------------------------------------------------------------
// Kernel 1: x_sq[b*N+n] = sum_c x[b,n,c]^2 ; one wave32 per row (64 floats ->
// float2 per lane, coalesced 256B row read), shuffle tree reduction.
// ---------------------------------------------------------------------------
__global__ void xsq_kernel(const float* __restrict__ x, float* __restrict__ xsq) {
    int gid  = blockIdx.x * blockDim.x + threadIdx.x;
    int wave = gid >> 5;              // row id, exact: grid = B*N/8 blocks * 8 waves
    int lane = threadIdx.x & 31;
    const float2* p = (const float2*)(x + (size_t)wave * C_);
    float2 v = p[lane];
    float s = v.x * v.x + v.y * v.y;
#pragma unroll
    for (int m = 16; m >= 1; m >>= 1) s += __shfl_xor(s, m, 32);
    if (lane == 0) xsq[wave] = s;
}

// ---------------------------------------------------------------------------
// Kernel 2: per-edge distance via V_WMMA_F32_16X16X4_F32.
// One wave = 16 edges. A = gathered xi rows (16x4 fp32 tile: lanes 0-15 hold
// K={0,1}, lanes 16-31 hold K={2,3}); B = gathered xj rows with the mirrored
// 4x16 layout. 16 chained WMMAs cover C=64; wanted dots = diagonal of the
// 16x16 f32 accumulator. Then dist = sqrt(relu(xsq_i+xsq_j-2*dot)+1e-6),
// summed over the 4 batches, /4, scatter atomicAdd into importance[row].
// EXEC is all-ones through every WMMA (exact grid, branchless diag extract).
// ---------------------------------------------------------------------------
__global__ void edge_kernel(const float* __restrict__ x,
                            const int*   __restrict__ row,
                            const int*   __restrict__ col,
                            const float* __restrict__ xsq,
                            float*       __restrict__ imp) {
    int gid   = blockIdx.x * blockDim.x + threadIdx.x;
    int wave  = gid >> 5;             // tile id, exact: E/16 waves total
    int lane  = threadIdx.x & 31;
    int ebase = wave << 4;
    int e16   = lane & 15;            // which of the 16 edges this lane serves
    int khalf = lane >> 4;            // K sub-block {0,1} vs {2,3}

    int r = row[ebase + e16];
    int c = col[ebase + e16];

    float dsum = 0.0f;
#pragma unroll
    for (int b = 0; b < B_; ++b) {
        const float* pr = x + ((size_t)b * N_ + (size_t)r) * C_ + khalf * 2;
        const float* pc = x + ((size_t)b * N_ + (size_t)c) * C_ + khalf * 2;
        v8f acc = {};
#pragma unroll
        for (int kb = 0; kb < 16; ++kb) {        // K = 64 in steps of 4
            v2f a, bb;
            a.x  = pr[kb * 4 + 0];  a.y  = pr[kb * 4 + 1];
            bb.x = pc[kb * 4 + 0];  bb.y = pc[kb * 4 + 1];
            // 8 args: (neg_a, A, neg_b, B, c_mod, C, reuse_a, reuse_b)
            acc = __builtin_amdgcn_wmma_f32_16x16x4_f32(
                false, a, false, bb, (short)0, acc, false, false);
        }
        // Diagonal extract: D[e][e]. e<8 -> lane e, vgpr e ; e>=8 -> lane e+8,
        // vgpr e-8. Each lane picks acc[lane&7] (cndmask chain), then shuffle.
        int i = lane & 7;
        float val = (i < 4) ? ((i < 2) ? (i == 0 ? acc[0] : acc[1])
                                       : (i == 2 ? acc[2] : acc[3]))
                            : ((i < 6) ? (i == 4 ? acc[4] : acc[5])
                                       : (i == 6 ? acc[6] : acc[7]));
        int srcl  = (e16 < 8) ? e16 : (e16 + 8);
        float dot = __shfl(val, srcl, 32);

        float d2 = xsq[b * N_ + r] + xsq[b * N_ + c] - 2.0f * dot;
        d2 = d2 > 0.0f ? d2 : 0.0f;
        dsum += sqrtf(d2 + 1e-6f);
    }
    if (lane < 16)                    // branch only after all WMMAs
        atomicAdd(&imp[r], dsum * 0.25f);
}

// ---------------------------------------------------------------------------
// Kernel 3: fused sigmoid weighting + k=2 average pool.
// out[b,m,c] = 0.5*(x[b,2m,c]*sig(imp[2m]) + x[b,2m+1,c]*sig(imp[2m+1]))
// ---------------------------------------------------------------------------
__global__ void pool_kernel(const float* __restrict__ x,
                            const float* __restrict__ imp,
                            float*       __restrict__ out) {
    int idx = blockIdx.x * blockDim.x + threadIdx.x;   // exact: B*NK*C threads
    int cch = idx & (C_ - 1);
    int m   = (idx >> 6) % NK_;
    int b   = idx / (NK_ * C_);
    int n0  = 2 * m, n1 = 2 * m + 1;
    float w0 = 1.0f / (1.0f + expf(-imp[n0]));
    float w1 = 1.0f / (1.0f + expf(-imp[n1]));
    float v0 = x[((size_t)b * N_ + n0) * C_ + cch];
    float v1 = x[((size_t)b * N_ + n1) * C_ + cch];
    out[idx] = 0.5f * (v0 * w0 + v1 * w1);
}

// ---------------------------------------------------------------------------
// Kernel 4: new_edge_index [2, 2*(NK-1)] flat row-major, written as floats:
// row0 = [0..M-1] ++ [1..M] ; row1 = [1..M] ++ [0..M-1], M = NK-1.
// ---------------------------------------------------------------------------
__global__ void eidx_kernel(float* __restrict__ out2) {
    int j = blockIdx.x * blockDim.x + threadIdx.x;
    const int M = NK_ - 1;
    if (j < M) {
        out2[j]         = (float)j;
        out2[M + j]     = (float)(j + 1);
        out2[2 * M + j] = (float)(j + 1);
        out2[3 * M + j] = (float)j;
    }
}

extern "C" void kernel_launch(void* const* d_in, const int* in_sizes, int n_in,
                              void* d_out, int out_size, void* d_ws, size_t ws_size,
                              hipStream_t stream) {
    const float* x  = (const float*)d_in[0];
    const int*   ei = (const int*)d_in[1];
    const int*   row = ei;          // edge_index[0]
    const int*   col = ei + E_;     // edge_index[1]
    float* out = (float*)d_out;

    float* xsq = (float*)d_ws;                     // B*N floats (320 KB)
    float* imp = xsq + (size_t)B_ * N_;            // N floats   (80 KB)

    // importance must be zeroed every call (async memset is graph-capturable)
    hipMemsetAsync(imp, 0, N_ * sizeof(float), stream);

    // 1 wave per row: B*N = 80000 rows, 8 waves per 256-thread block
    xsq_kernel<<<(B_ * N_) / 8, 256, 0, stream>>>(x, xsq);

    // 16 edges per wave: E/16 = 20000 waves, 8 per block
    edge_kernel<<<E_ / 128, 256, 0, stream>>>(x, row, col, xsq, imp);

    // one thread per output element: 4*10000*64 = 2,560,000
    pool_kernel<<<(B_ * NK_ * C_) / 256, 256, 0, stream>>>(x, imp, out);

    // chain edge index appended after the pooled block
    eidx_kernel<<<(NK_ - 1 + 255) / 256, 256, 0, stream>>>(out + (size_t)B_ * NK_ * C_);
}